// KANLayer_65163243815210
// MI455X (gfx1250) — compile-verified
//
#include <hip/hip_runtime.h>

typedef __attribute__((ext_vector_type(2))) float v2f;
typedef __attribute__((ext_vector_type(8))) float v8f;

#define DIN   64
#define DOUT  64
#define HID   32
#define BATCH 1024

// Grid: 64 blocks (one per 16-row batch tile), 256 threads = 8 waves.
// Wave w handles output columns o in [w*8, w*8+8), loops over all i=0..63.
__global__ __launch_bounds__(256) void kan_wmma_kernel(
    const float* __restrict__ x,    // [B, DIN]
    const float* __restrict__ W1,   // [DIN, DOUT, HID, 7]
    const float* __restrict__ W2,   // [DIN, DOUT, 1, HID]
    const float* __restrict__ B1,   // [DIN, DOUT, HID]
    const float* __restrict__ B2,   // [DIN, DOUT, 1]
    float* __restrict__ out)        // [B, DOUT]
{
    const int lane  = threadIdx.x & 31;
    const int wave  = threadIdx.x >> 5;   // 0..7
    const int btile = blockIdx.x;         // 0..63
    const int col   = lane & 15;          // N (and M-low) index
    const bool hi   = lane >= 16;
    const int koff  = hi ? 2 : 0;         // K base for this half-wave
    const int obase = wave * 8;

    // Per-lane accumulators: acc[oo][v] is partial sum for row M=v(+8 if hi),
    // column-group N=col, output o=obase+oo, summed over i and both h-tiles.
    float acc[8][8];
    float bsum[8];
#pragma unroll
    for (int oo = 0; oo < 8; ++oo) {
        bsum[oo] = 0.0f;
#pragma unroll
        for (int v = 0; v < 8; ++v) acc[oo][v] = 0.0f;
    }

    for (int i = 0; i < DIN; ++i) {
        // ---- A operand: feats[b, f] for b = btile*16 + col ----
        const float xv = x[(btile * 16 + col) * DIN + i];
        const float s1 = __sinf(2.0f * xv);
        const float s2 = __sinf(4.0f * xv);
        const float s3 = __sinf(8.0f * xv);
        const float c1 = __cosf(2.0f * xv);
        const float c2 = __cosf(4.0f * xv);
        const float c3 = __cosf(8.0f * xv);
        // feats = [x, sin2x, sin4x, sin8x, cos2x, cos4x, cos8x, 0]
        v2f a0, a1;                         // K-tiles 0..3 and 4..7
        a0.x = hi ? s2 : xv;                // K = koff+0  -> f2 / f0
        a0.y = hi ? s3 : s1;                // K = koff+1  -> f3 / f1
        a1.x = hi ? c3 : c1;                // K = 4+koff  -> f6 / f4
        a1.y = hi ? 0.0f : c2;              // K = 5+koff  -> f7(pad) / f5

        const float* W1i = W1 + (size_t)i * DOUT * HID * 7;
        const float* B1i = B1 + (size_t)i * DOUT * HID;
        const float* W2i = W2 + (size_t)i * DOUT * HID;
        const float* B2i = B2 + (size_t)i * DOUT;

#pragma unroll
        for (int oo = 0; oo < 8; ++oo) {
            const int o = obase + oo;
            bsum[oo] += B2i[o];
            const float* w1o = W1i + (size_t)o * HID * 7;
            const float* b1o = B1i + (size_t)o * HID;
            const float* w2o = W2i + (size_t)o * HID;

#pragma unroll
            for (int ht = 0; ht < 2; ++ht) {
                const int hidx = ht * 16 + col;       // N -> hidden unit
                const float* p = w1o + hidx * 7;

                // ---- B operand: W1[i,o,hidx,f], f=7 zero-padded ----
                v2f b0, b1v;
                b0.x  = p[koff];                      // f = koff
                b0.y  = p[koff + 1];                  // f = koff+1
                b1v.x = p[4 + koff];                  // f = 4+koff (f6 for hi)
                b1v.y = hi ? 0.0f : p[5];             // f = 7 pad / f5

                const float bias = b1o[hidx];
                const float w2v  = w2o[hidx];

                // ---- C seeded with bias (broadcast over batch rows) ----
                v8f c;
#pragma unroll
                for (int v = 0; v < 8; ++v) c[v] = bias;

                // D = feats * W1^T + bias   (K = 8, two chained K=4 WMMAs)
                c = __builtin_amdgcn_wmma_f32_16x16x4_f32(
                        false, a1, false, b1v, (short)0, c, false, false);
                c = __builtin_amdgcn_wmma_f32_16x16x4_f32(
                        false, a0, false, b0, (short)0, c, false, false);

                // ---- SiLU + second-layer scale, accumulate ----
                // silu(h) = h * 1/(1+exp(-h)); use native v_rcp_f32 (1 ulp)
                // instead of the IEEE div sequence (div_scale/div_fixup).
#pragma unroll
                for (int v = 0; v < 8; ++v) {
                    const float hh  = c[v];
                    const float sig = __builtin_amdgcn_rcpf(1.0f + __expf(-hh));
                    acc[oo][v] += (hh * sig) * w2v;
                }
            }
        }
    }

    // ---- Deferred N-reduction (sum over 16 lanes of each half-wave) ----
#pragma unroll
    for (int oo = 0; oo < 8; ++oo) {
        const int o = obase + oo;
#pragma unroll
        for (int v = 0; v < 8; ++v) {
            float r = acc[oo][v];
            r += __shfl_xor(r, 1);
            r += __shfl_xor(r, 2);
            r += __shfl_xor(r, 4);
            r += __shfl_xor(r, 8);
            if (col == 0) {
                const int m = v + (hi ? 8 : 0);       // batch row within tile
                out[(btile * 16 + m) * DOUT + o] = r + bsum[oo];
            }
        }
    }
}

extern "C" void kernel_launch(void* const* d_in, const int* in_sizes, int n_in,
                              void* d_out, int out_size, void* d_ws, size_t ws_size,
                              hipStream_t stream) {
    const float* x  = (const float*)d_in[0];
    const float* W1 = (const float*)d_in[1];
    const float* W2 = (const float*)d_in[2];
    const float* B1 = (const float*)d_in[3];
    const float* B2 = (const float*)d_in[4];
    float* out = (float*)d_out;

    (void)in_sizes; (void)n_in; (void)out_size; (void)d_ws; (void)ws_size;

    kan_wmma_kernel<<<dim3(BATCH / 16), dim3(256), 0, stream>>>(
        x, W1, W2, B1, B2, out);
}